// Decoder_KPA_40862318854673
// MI455X (gfx1250) — compile-verified
//
#include <hip/hip_runtime.h>
#include <math.h>
#include <stdint.h>

typedef float v2f __attribute__((ext_vector_type(2)));
typedef float v4f __attribute__((ext_vector_type(4)));
typedef float v8f __attribute__((ext_vector_type(8)));
typedef int   v4i_g __attribute__((vector_size(4 * sizeof(int))));   // gcc-style, matches builtin proto

#if defined(__has_builtin)
#if __has_builtin(__builtin_amdgcn_global_load_async_to_lds_b128)
#define KPA_HAS_ASYNC_LDS 1
#endif
#if __has_builtin(__builtin_amdgcn_s_wait_asynccnt)
#define KPA_HAS_WAIT_ASYNC 1
#endif
#endif

__device__ __forceinline__ float gelu_exact(float x) {
    // jax.nn.gelu(approximate=False) = 0.5*x*(1+erf(x/sqrt(2)))
    return 0.5f * x * (1.0f + erff(x * 0.70710678118654752440f));
}

// ---------------------------------------------------------------------------
// Adaptive 3x3 average pool (torch bin rule, overlapping bins) of y_m and y_w
// into featT rows. featT layout: (48, 2C) row-major; row = b*10+pos,
// pos 0..8 = 3x3 bins (kh*3+kw), pos 9 = mean of the 9 pooled values.
// Channel ch < C -> pooled y_m (moire); ch >= C -> pooled y_w.
// One block per (b, ch); 256 threads.
// ---------------------------------------------------------------------------
__global__ void kpa_pool_kernel(const float* __restrict__ ym,
                                const float* __restrict__ yw,
                                float* __restrict__ featT,
                                int C, int H, int W) {
    const int twoC = 2 * C;
    const int blk = blockIdx.x;
    const int b = blk / twoC;
    const int ch = blk % twoC;
    const float* plane = (ch < C)
        ? (ym + (size_t)(b * C + ch) * H * W)
        : (yw + (size_t)(b * C + (ch - C)) * H * W);

    int h0[3], h1[3], w0[3], w1[3];
#pragma unroll
    for (int j = 0; j < 3; ++j) {
        h0[j] = (j * H) / 3;  h1[j] = ((j + 1) * H + 2) / 3;   // ceil
        w0[j] = (j * W) / 3;  w1[j] = ((j + 1) * W + 2) / 3;
    }

    float acc[9];
#pragma unroll
    for (int i = 0; i < 9; ++i) acc[i] = 0.0f;

    const int tid = threadIdx.x;
    const int n = H * W;
    for (int idx = tid; idx < n; idx += blockDim.x) {
        const int h = idx / W, w = idx % W;
        const float v = plane[idx];
#pragma unroll
        for (int j = 0; j < 3; ++j) {
            if (h >= h0[j] && h < h1[j]) {
#pragma unroll
                for (int i = 0; i < 3; ++i) {
                    if (w >= w0[i] && w < w1[i]) acc[3 * j + i] += v;
                }
            }
        }
    }

    __shared__ float sbin[9];
    __shared__ float spool[9];
    if (tid < 9) sbin[tid] = 0.0f;
    __syncthreads();
#pragma unroll
    for (int i = 0; i < 9; ++i) atomicAdd(&sbin[i], acc[i]);   // ds_add_f32
    __syncthreads();
    if (tid < 9) {
        const int j = tid / 3, i = tid % 3;
        const float area = (float)((h1[j] - h0[j]) * (w1[i] - w0[i]));
        const float p = sbin[tid] / area;
        spool[tid] = p;
        featT[(size_t)(b * 10 + tid) * twoC + ch] = p;
    }
    __syncthreads();
    if (tid == 0) {
        float m = 0.0f;
#pragma unroll
        for (int i = 0; i < 9; ++i) m += spool[i];
        featT[(size_t)(b * 10 + 9) * twoC + ch] = m * (1.0f / 9.0f);
    }
}

// Zero the M-padding rows (40..47) of featT so WMMA A-tile loads need no
// predication (keeps EXEC untouched around the WMMA loop).
__global__ void kpa_pad_zero_kernel(float* __restrict__ featT, int twoC) {
    const int t = blockIdx.x * blockDim.x + threadIdx.x;
    const int n = 8 * twoC;
    if (t < n) featT[(size_t)40 * twoC + t] = 0.0f;
}

// ---------------------------------------------------------------------------
// D(48, N) = [gelu]( A(48,K) @ Wt(N,K)^T + bias(N) )   via WMMA f32 16x16x4.
// One wave per 16x16 output tile; grid.x = 3 * Ntiles. A rows 40..47 are
// zero-padded by kpa_pad_zero_kernel (stage 1) or written-but-unused (stage 2).
// A-tile (16 rows x K) and B-tile (16 Wt rows x K) are both contiguous 16*K
// float blocks -> staged into LDS with GLOBAL_LOAD_ASYNC_TO_LDS_B128
// (ASYNCcnt) when the toolchain exposes the builtin, else a plain LDS copy.
// Fragment layouts per CDNA5 ISA 7.12.2:
//   A 16x4:  lane holds row M=lane&15; VGPR0/1 = K,K+1; lanes>=16 use K+2,K+3
//   B 4x16:  VGPR holds K-row striped across lanes; lanes>=16 hold K+2 half
//   C/D:     VGPR v -> M = v + 8*(lane>=16), N = lane&15
// ---------------------------------------------------------------------------
template <int K>
__global__ void kpa_mlp_kernel(const float* __restrict__ A,   // (48, K)
                               const float* __restrict__ Wt,  // (N, K)
                               const float* __restrict__ bias,
                               float* __restrict__ D,         // (48, N)
                               int N, int apply_gelu, int Ntiles) {
    __shared__ __align__(16) float smem[2 * 16 * K];
    float* sA = smem;
    float* sB = smem + 16 * K;

    const int lane = threadIdx.x;
    const int mt = blockIdx.x / Ntiles;
    const int nt = blockIdx.x % Ntiles;
    const int l15 = lane & 15;
    const int hi  = lane >> 4;          // 0 or 1

    const float* gA = A  + (size_t)mt * 16 * K;   // contiguous 16*K floats
    const float* gB = Wt + (size_t)nt * 16 * K;   // contiguous 16*K floats
    const int tileFloats = 16 * K;

#if defined(KPA_HAS_ASYNC_LDS)
    typedef __attribute__((address_space(1))) v4i_g gv4i;   // global int32x4
    typedef __attribute__((address_space(3))) v4i_g lv4i;   // LDS int32x4
    for (int off = lane * 4; off < tileFloats; off += 128) {   // b128 per lane
        __builtin_amdgcn_global_load_async_to_lds_b128(
            (gv4i*)(uintptr_t)(gA + off),
            (lv4i*)(uint32_t)(uintptr_t)(sA + off), 0, 0);
        __builtin_amdgcn_global_load_async_to_lds_b128(
            (gv4i*)(uintptr_t)(gB + off),
            (lv4i*)(uint32_t)(uintptr_t)(sB + off), 0, 0);
    }
#if defined(KPA_HAS_WAIT_ASYNC)
    __builtin_amdgcn_s_wait_asynccnt(0);
#else
    asm volatile("s_wait_asynccnt 0x0" ::: "memory");
#endif
#else
    for (int off = lane; off < tileFloats; off += 32) {
        sA[off] = gA[off];
        sB[off] = gB[off];
    }
    // single wave: LDS ops are in-order within a wave, no barrier needed
#endif

    v8f acc = {0.f, 0.f, 0.f, 0.f, 0.f, 0.f, 0.f, 0.f};
#pragma unroll 4
    for (int k = 0; k < K; k += 4) {
        const int ka = k + (hi << 1);
        v2f a, bf;
        a.x  = sA[l15 * K + ka];
        a.y  = sA[l15 * K + ka + 1];
        bf.x = sB[l15 * K + ka];
        bf.y = sB[l15 * K + ka + 1];
        acc = __builtin_amdgcn_wmma_f32_16x16x4_f32(
            /*neg_a=*/false, a, /*neg_b=*/false, bf,
            /*c_mod=*/(short)0, acc, /*reuse_a=*/false, /*reuse_b=*/false);
    }

    const float bn = bias[nt * 16 + l15];
#pragma unroll
    for (int v = 0; v < 8; ++v) {
        const int mo = mt * 16 + hi * 8 + v;
        float val = acc[v] + bn;
        if (apply_gelu) val = gelu_exact(val);
        D[(size_t)mo * N + nt * 16 + l15] = val;
    }
}

// ---------------------------------------------------------------------------
// Per (b,c): softmax over the G=2 group logits, blend Wp / bias_p.
// S layout (48, 2C), valid rows m = b*10 + pos (pos 0..8 spatial, 9 = mean).
// ---------------------------------------------------------------------------
__global__ void kpa_weights_kernel(const float* __restrict__ S,
                                   const float* __restrict__ Wp,   // (2, C, 9)
                                   const float* __restrict__ bp,   // (2, C)
                                   float* __restrict__ wout,       // (B*C, 9)
                                   float* __restrict__ bout,       // (B*C)
                                   int B, int C) {
    const int t = blockIdx.x * blockDim.x + threadIdx.x;
    if (t >= B * C) return;
    const int b = t / C, c = t % C;
    const int twoC = 2 * C;
#pragma unroll
    for (int pos = 0; pos < 9; ++pos) {
        const float* row = S + (size_t)(b * 10 + pos) * twoC;
        const float s0 = row[c], s1 = row[C + c];
        const float w0 = 1.0f / (1.0f + expf(s1 - s0));
        const float w1 = 1.0f - w0;
        wout[(size_t)t * 9 + pos] =
            w0 * Wp[c * 9 + pos] + w1 * Wp[(C + c) * 9 + pos];
    }
    const float* rowm = S + (size_t)(b * 10 + 9) * twoC;
    const float s0 = rowm[c], s1 = rowm[C + c];
    const float w0 = 1.0f / (1.0f + expf(s1 - s0));
    bout[t] = w0 * bp[c] + (1.0f - w0) * bp[C + c];
}

// ---------------------------------------------------------------------------
// out = y_m + alpha * (depthwise3x3(y_w, wgt[b,c]) + bias[b,c])
// Memory-bound streaming kernel: 4 output pixels per thread, float4 y_m read
// and float4 store; zero padding at image borders.
// ---------------------------------------------------------------------------
__global__ void kpa_conv_kernel(const float* __restrict__ yw,
                                const float* __restrict__ ym,
                                const float* __restrict__ wgt,   // (B*C, 9)
                                const float* __restrict__ bias,  // (B*C)
                                const float* __restrict__ alpha_p,
                                float* __restrict__ out,
                                int H, int W, int total_quads) {
    const int t = blockIdx.x * blockDim.x + threadIdx.x;
    if (t >= total_quads) return;
    const int Wq = W >> 2;
    const int wq = t % Wq;
    const int rr = t / Wq;
    const int h  = rr % H;
    const int bc = rr / H;            // b*C + c

    const size_t pbase = (size_t)bc * H * W;
    const float* pw = yw + pbase;
    const float* pm = ym + pbase;
    float*       po = out + pbase;

    float kv[9];
#pragma unroll
    for (int i = 0; i < 9; ++i) kv[i] = wgt[(size_t)bc * 9 + i];
    const float bi = bias[bc];
    const float alpha = *alpha_p;

    const int wbase = wq << 2;
    float o0 = 0.f, o1 = 0.f, o2 = 0.f, o3 = 0.f;

#pragma unroll
    for (int dh = -1; dh <= 1; ++dh) {
        const int hr = h + dh;
        if (hr < 0 || hr >= H) continue;
        const float* rowp = pw + (size_t)hr * W;
        float v[6];
#pragma unroll
        for (int j = 0; j < 6; ++j) {
            const int wc = wbase - 1 + j;
            v[j] = (wc >= 0 && wc < W) ? rowp[wc] : 0.0f;
        }
        const float ka = kv[(dh + 1) * 3 + 0];
        const float kb = kv[(dh + 1) * 3 + 1];
        const float kc = kv[(dh + 1) * 3 + 2];
        o0 += v[0] * ka + v[1] * kb + v[2] * kc;
        o1 += v[1] * ka + v[2] * kb + v[3] * kc;
        o2 += v[2] * ka + v[3] * kb + v[4] * kc;
        o3 += v[3] * ka + v[4] * kb + v[5] * kc;
    }

    // Pull the next chunk of this plane toward the caches (global_prefetch_b8).
    const size_t pfoff = (size_t)h * W + wbase + 2048;
    if (pfoff < (size_t)H * W) __builtin_prefetch(pw + pfoff, 0, 0);

    const size_t off = (size_t)h * W + wbase;
    const v4f m = *(const v4f*)(pm + off);
    v4f res;
    res.x = m.x + alpha * (o0 + bi);
    res.y = m.y + alpha * (o1 + bi);
    res.z = m.z + alpha * (o2 + bi);
    res.w = m.w + alpha * (o3 + bi);
    *(v4f*)(po + off) = res;
}

// ---------------------------------------------------------------------------
// Host-side orchestration. Stream-ordered launches; one scratch region reused
// across the three scales (the stream serializes the dependency chain).
// ---------------------------------------------------------------------------
extern "C" void kernel_launch(void* const* d_in, const int* in_sizes, int n_in,
                              void* d_out, int out_size, void* d_ws, size_t ws_size,
                              hipStream_t stream) {
    (void)in_sizes; (void)n_in; (void)out_size; (void)ws_size;
    const int B = 4;
    const int Cs[3] = {64, 128, 256};
    const int Hs[3] = {256, 128, 64};

    const float* ym[3] = {(const float*)d_in[0], (const float*)d_in[1], (const float*)d_in[2]};
    const float* yw[3] = {(const float*)d_in[3], (const float*)d_in[4], (const float*)d_in[5]};
    const float* al[3] = {(const float*)d_in[6], (const float*)d_in[7], (const float*)d_in[8]};

    float* ws   = (float*)d_ws;
    float* outp = (float*)d_out;

    // scratch layout (floats), sized for the largest scale (C=256):
    float* featT = ws;            // (48, 2C)        <= 24576
    float* Hbuf  = ws + 24576;    // (48, r)         <=  3072
    float* Sbuf  = ws + 28672;    // (48, 2C)        <= 24576
    float* wgt   = ws + 53248;    // (B*C, 9)        <=  9216
    float* bvec  = ws + 62464;    // (B*C)           <=  1024

    size_t out_off = 0;
    for (int s = 0; s < 3; ++s) {
        const int C = Cs[s], H = Hs[s], W = Hs[s];
        const int twoC = 2 * C, r = C / 4;
        const float* Wp  = (const float*)d_in[9 + 6 * s + 0];
        const float* bp  = (const float*)d_in[9 + 6 * s + 1];
        const float* p1w = (const float*)d_in[9 + 6 * s + 2];
        const float* p1b = (const float*)d_in[9 + 6 * s + 3];
        const float* p2w = (const float*)d_in[9 + 6 * s + 4];
        const float* p2b = (const float*)d_in[9 + 6 * s + 5];

        // 1) adaptive 3x3 pool of (moire, x) + per-channel mean row
        kpa_pool_kernel<<<dim3(B * twoC), dim3(256), 0, stream>>>(
            ym[s], yw[s], featT, C, H, W);
        // zero M-pad rows 40..47 so WMMA A loads are unpredicated
        kpa_pad_zero_kernel<<<dim3((8 * twoC + 255) / 256), dim3(256), 0, stream>>>(
            featT, twoC);

        // 2) H = gelu(feat @ p1w^T + p1b)     (WMMA f32 16x16x4, K = 2C)
        const int nt1 = r / 16;
        switch (twoC) {
        case 128:
            kpa_mlp_kernel<128><<<dim3(3 * nt1), dim3(32), 0, stream>>>(
                featT, p1w, p1b, Hbuf, r, 1, nt1); break;
        case 256:
            kpa_mlp_kernel<256><<<dim3(3 * nt1), dim3(32), 0, stream>>>(
                featT, p1w, p1b, Hbuf, r, 1, nt1); break;
        default:
            kpa_mlp_kernel<512><<<dim3(3 * nt1), dim3(32), 0, stream>>>(
                featT, p1w, p1b, Hbuf, r, 1, nt1); break;
        }

        // 3) S = H @ p2w^T + p2b              (WMMA f32 16x16x4, K = r)
        const int nt2 = twoC / 16;
        switch (r) {
        case 16:
            kpa_mlp_kernel<16><<<dim3(3 * nt2), dim3(32), 0, stream>>>(
                Hbuf, p2w, p2b, Sbuf, twoC, 0, nt2); break;
        case 32:
            kpa_mlp_kernel<32><<<dim3(3 * nt2), dim3(32), 0, stream>>>(
                Hbuf, p2w, p2b, Sbuf, twoC, 0, nt2); break;
        default:
            kpa_mlp_kernel<64><<<dim3(3 * nt2), dim3(32), 0, stream>>>(
                Hbuf, p2w, p2b, Sbuf, twoC, 0, nt2); break;
        }

        // 4) group softmax -> per-(b,c) 3x3 kernel + bias
        kpa_weights_kernel<<<dim3((B * C + 255) / 256), dim3(256), 0, stream>>>(
            Sbuf, Wp, bp, wgt, bvec, B, C);

        // 5) out = y_m + a * (depthwise conv(y_w) + bias)
        const int total_quads = B * C * H * (W / 4);
        kpa_conv_kernel<<<dim3((total_quads + 255) / 256), dim3(256), 0, stream>>>(
            yw[s], ym[s], wgt, bvec, al[s], outp + out_off, H, W, total_quads);

        out_off += (size_t)B * C * H * W;
    }
}